// GraphUnetLayer_83631603187833
// MI455X (gfx1250) — compile-verified
//
#include <hip/hip_runtime.h>
#include <cstdint>

// ---------------------------------------------------------------------------
// Graph U-Net (N=100000, E=1000000, D=64, L=3, rates 0.5) for gfx1250.
// Dense 64x64 GEMMs use V_WMMA_F32_16X16X4_F32 (exact fp32 matrix path).
// SpMM is one wave32 per edge with float2 coalesced gathers + f32 atomics.
// Top-k replicated exactly via stable 64-bit LSD radix sort.
// ---------------------------------------------------------------------------

typedef float v2f __attribute__((ext_vector_type(2)));
typedef float v8f __attribute__((ext_vector_type(8)));

#define GUN_N 100000
#define GUN_E 1000000
#define GUN_D 64
#define GUN_L 3

// ---------------- elementwise helpers ----------------
__global__ void fill_f32(float* __restrict__ p, float v, int n) {
  int i = blockIdx.x * blockDim.x + threadIdx.x;
  if (i < n) p[i] = v;
}
__global__ void fill_i32(int* __restrict__ p, int v, int n) {
  int i = blockIdx.x * blockDim.x + threadIdx.x;
  if (i < n) p[i] = v;
}
__global__ void elu_k(const float* __restrict__ a, float* __restrict__ o, int n) {
  int i = blockIdx.x * blockDim.x + threadIdx.x;
  if (i < n) { float x = a[i]; o[i] = x > 0.0f ? x : expm1f(x); }
}
__global__ void elu_add_k(const float* __restrict__ a, const float* __restrict__ skip,
                          float* __restrict__ o, int n) {
  int i = blockIdx.x * blockDim.x + threadIdx.x;
  if (i < n) { float x = a[i]; x = x > 0.0f ? x : expm1f(x); o[i] = x + skip[i]; }
}
__global__ void add_k(const float* __restrict__ a, const float* __restrict__ b,
                      float* __restrict__ o, int n) {
  int i = blockIdx.x * blockDim.x + threadIdx.x;
  if (i < n) o[i] = a[i] + b[i];
}

// ---------------- WMMA fp32 GEMM: out = A(n x 64) @ W(64 x 64) + bias ----------------
// Block = 128 threads (4 waves). Wave w computes the 16x16 tile at cols [16w,16w+16),
// rows [16*blockIdx.x, +16). K=64 -> 16 chained v_wmma_f32_16x16x4_f32.
__global__ void __launch_bounds__(128)
gemm64_wmma(const float* __restrict__ A, const float* __restrict__ W,
            const float* __restrict__ bias, float* __restrict__ out, int n) {
  const int wave = threadIdx.x >> 5;
  const int lane = threadIdx.x & 31;
  const int lm   = lane & 15;            // row-within-tile for A, col for B/C
  const int koff = (lane >> 4) << 1;     // lanes 16-31 hold K+2,K+3
  const int m0   = blockIdx.x << 4;
  const int col  = (wave << 4) + lm;     // 0..63

  int mrow = m0 + lm;
  if (mrow >= n) mrow = n - 1;           // clamp (stores are guarded); EXEC stays full
  const float* arow = A + (size_t)mrow * GUN_D;

  v8f c = {0.f, 0.f, 0.f, 0.f, 0.f, 0.f, 0.f, 0.f};
#pragma unroll
  for (int k0 = 0; k0 < GUN_D; k0 += 4) {
    v2f a = *(const v2f*)(arow + k0 + koff);
    v2f b;
    b.x = W[(k0 + koff) * GUN_D + col];
    b.y = W[(k0 + koff + 1) * GUN_D + col];
    c = __builtin_amdgcn_wmma_f32_16x16x4_f32(false, a, false, b, (short)0, c,
                                              false, false);
  }
  const float bv = bias[col];
  const int mbase = m0 + ((lane >> 4) << 3);
#pragma unroll
  for (int r = 0; r < 8; ++r) {
    int row = mbase + r;
    if (row < n) out[(size_t)row * GUN_D + col] = c[r] + bv;
  }
}

// ---------------- SpMM: out[row[e]] += w[e] * hw[col[e]]  (one wave per edge) -------
__global__ void spmm_atomic(const int* __restrict__ row, const int* __restrict__ col,
                            const float* __restrict__ w, const float* __restrict__ hw,
                            float* __restrict__ out, int ne) {
  int e = blockIdx.x * (blockDim.x >> 5) + (threadIdx.x >> 5);
  if (e >= ne) return;
  float wt = w[e];
  if (wt == 0.0f) return;
  int lane = threadIdx.x & 31;
  int r = row[e], cix = col[e];
  const float2 v = *(const float2*)(hw + (size_t)cix * GUN_D + lane * 2);
  float* o = out + (size_t)r * GUN_D + lane * 2;
  atomicAdd(o,     wt * v.x);
  atomicAdd(o + 1, wt * v.y);
}

// ---------------- scores = tanh(h @ pw + pb[bi])  (one wave per row) ----------------
__global__ void score_k(const float* __restrict__ h, const float* __restrict__ pw,
                        const float* __restrict__ pb, int bi,
                        float* __restrict__ sc, int n) {
  int r = blockIdx.x * (blockDim.x >> 5) + (threadIdx.x >> 5);
  if (r >= n) return;
  int lane = threadIdx.x & 31;
  float2 hv = *(const float2*)(h + (size_t)r * GUN_D + lane * 2);
  float2 wv = *(const float2*)(pw + lane * 2);
  float p = hv.x * wv.x + hv.y * wv.y;
#pragma unroll
  for (int off = 16; off; off >>= 1) p += __shfl_xor(p, off, 32);
  if (lane == 0) sc[r] = tanhf(p + pb[bi]);
}

// ---------------- top-k via stable LSD radix sort on 64-bit keys --------------------
__global__ void build_keys(const float* __restrict__ sc,
                           unsigned long long* __restrict__ keys, int n) {
  int i = blockIdx.x * blockDim.x + threadIdx.x;
  if (i >= n) return;
  unsigned int u = __float_as_uint(sc[i]);
  u = (u & 0x80000000u) ? ~u : (u | 0x80000000u);     // ascending-orderable
  unsigned int ku = ~u;                               // flip -> descending score
  keys[i] = ((unsigned long long)ku << 32) | (unsigned int)i;
}

__global__ void radix_count(const unsigned long long* __restrict__ keys, int n,
                            unsigned int* __restrict__ hist, int nb, int shift) {
  __shared__ unsigned int lh[256];
  lh[threadIdx.x] = 0;
  __syncthreads();
  int i = blockIdx.x * 256 + threadIdx.x;
  if (i < n) {
    unsigned int d = (unsigned int)(keys[i] >> shift) & 255u;
    atomicAdd(&lh[d], 1u);
  }
  __syncthreads();
  hist[threadIdx.x * nb + blockIdx.x] = lh[threadIdx.x];
}

__global__ void radix_scan(unsigned int* __restrict__ hist, int nb) {
  __shared__ unsigned int totals[256];
  unsigned int t = 0;
  for (int b = 0; b < nb; ++b) t += hist[threadIdx.x * nb + b];
  totals[threadIdx.x] = t;
  __syncthreads();
  if (threadIdx.x == 0) {
    unsigned int run = 0;
    for (int d = 0; d < 256; ++d) { unsigned int v = totals[d]; totals[d] = run; run += v; }
  }
  __syncthreads();
  unsigned int run = totals[threadIdx.x];
  for (int b = 0; b < nb; ++b) {
    unsigned int v = hist[threadIdx.x * nb + b];
    hist[threadIdx.x * nb + b] = run;
    run += v;
  }
}

__global__ void radix_scatter(const unsigned long long* __restrict__ in,
                              unsigned long long* __restrict__ out, int n,
                              const unsigned int* __restrict__ hist, int nb, int shift) {
  __shared__ unsigned int digs[256];
  __shared__ unsigned int base[256];
  int i = blockIdx.x * 256 + threadIdx.x;
  bool valid = (i < n);
  unsigned long long k = 0;
  unsigned int d = 0;
  if (valid) { k = in[i]; d = (unsigned int)(k >> shift) & 255u; }
  digs[threadIdx.x] = valid ? d : 0xFFFFFFFFu;
  base[threadIdx.x] = hist[threadIdx.x * nb + blockIdx.x];
  __syncthreads();
  if (valid) {
    unsigned int rank = 0;
    for (int j = 0; j < threadIdx.x; ++j) rank += (digs[j] == d);
    out[base[d] + rank] = k;
  }
}

__global__ void extract_idx(const unsigned long long* __restrict__ keys,
                            int* __restrict__ idx, int k) {
  int i = blockIdx.x * blockDim.x + threadIdx.x;
  if (i < k) idx[i] = (int)(keys[i] & 0xFFFFFFFFull);
}

// ---------------- pooling graph rebuild ----------------
__global__ void set_newid(const int* __restrict__ idx, int k, int* __restrict__ newid) {
  int i = blockIdx.x * blockDim.x + threadIdx.x;
  if (i < k) newid[idx[i]] = i;
}

__global__ void edge_remap(const int* __restrict__ rp, const int* __restrict__ cp,
                           const int* __restrict__ newid, int ne,
                           int* __restrict__ ro, int* __restrict__ co,
                           float* __restrict__ wo, float* __restrict__ deg) {
  int e = blockIdx.x * blockDim.x + threadIdx.x;
  if (e >= ne) return;
  int nr = newid[rp[e]];
  int nc = newid[cp[e]];
  bool valid = (nr >= 0) && (nc >= 0);
  int nr2 = valid ? nr : 0;
  int nc2 = valid ? nc : 0;
  ro[e] = nr2;
  co[e] = nc2;
  wo[e] = valid ? 1.0f : 0.0f;
  if (valid) atomicAdd(&deg[nr2], 1.0f);
}

__global__ void ew_final(float* __restrict__ wo, const int* __restrict__ ro,
                         const float* __restrict__ deg, int ne) {
  int e = blockIdx.x * blockDim.x + threadIdx.x;
  if (e >= ne) return;
  wo[e] = wo[e] * (1.0f / fmaxf(deg[ro[e]], 1.0f));
}

// ---------------- row gather / scatter (pool / unpool) ----------------
__global__ void gather_rows(const float* __restrict__ src, const int* __restrict__ idx,
                            float* __restrict__ dst, int cnt) {
  int t = blockIdx.x * blockDim.x + threadIdx.x;
  if (t >= cnt) return;
  int j = t >> 6, f = t & 63;
  dst[t] = src[(size_t)idx[j] * GUN_D + f];
}
__global__ void scatter_rows(const float* __restrict__ src, const int* __restrict__ idx,
                             float* __restrict__ dst, int cnt) {
  int t = blockIdx.x * blockDim.x + threadIdx.x;
  if (t >= cnt) return;
  int j = t >> 6, f = t & 63;
  dst[(size_t)idx[j] * GUN_D + f] = src[t];
}

// ---------------------------------------------------------------------------
extern "C" void kernel_launch(void* const* d_in, const int* in_sizes, int n_in,
                              void* d_out, int out_size, void* d_ws, size_t ws_size,
                              hipStream_t stream) {
  (void)in_sizes; (void)n_in; (void)out_size; (void)ws_size;
  constexpr int N = GUN_N, E = GUN_E, D = GUN_D, L = GUN_L;
  const int sizes[4] = {100000, 50000, 25000, 12500};

  const float* x      = (const float*)d_in[0];
  const int*   erow0  = (const int*)d_in[1];
  const int*   ecol0  = erow0 + E;
  const float* ew0    = (const float*)d_in[2];
  const float* gcn_w  = (const float*)d_in[3];
  const float* gcn_b  = (const float*)d_in[4];
  const float* down_w = (const float*)d_in[5];
  const float* down_b = (const float*)d_in[6];
  const float* up_w   = (const float*)d_in[7];
  const float* up_b   = (const float*)d_in[8];
  const float* proj_w = (const float*)d_in[9];
  const float* proj_b = (const float*)d_in[10];

  // bump allocator over d_ws
  char* bp = (char*)d_ws;
  auto alloc = [&](size_t bytes) -> void* {
    void* r = (void*)bp;
    bp += (bytes + 255) & ~(size_t)255;
    return r;
  };
  float* HW   = (float*)alloc(sizeof(float) * (size_t)N * D);
  float* ACC  = (float*)alloc(sizeof(float) * (size_t)N * D);   // also unpool target
  float* HD0  = (float*)alloc(sizeof(float) * (size_t)100000 * D);
  float* HD1  = (float*)alloc(sizeof(float) * (size_t)50000 * D);
  float* HD2  = (float*)alloc(sizeof(float) * (size_t)25000 * D);
  float* HCUR = (float*)alloc(sizeof(float) * (size_t)50000 * D);
  int*   row1 = (int*)alloc(sizeof(int) * E);
  int*   col1 = (int*)alloc(sizeof(int) * E);
  float* ewl1 = (float*)alloc(sizeof(float) * E);
  int*   row2 = (int*)alloc(sizeof(int) * E);
  int*   col2 = (int*)alloc(sizeof(int) * E);
  float* ewl2 = (float*)alloc(sizeof(float) * E);
  int*   row3 = (int*)alloc(sizeof(int) * E);
  int*   col3 = (int*)alloc(sizeof(int) * E);
  float* ewl3 = (float*)alloc(sizeof(float) * E);
  float* scores = (float*)alloc(sizeof(float) * N);
  unsigned long long* keyA = (unsigned long long*)alloc(sizeof(unsigned long long) * N);
  unsigned long long* keyB = (unsigned long long*)alloc(sizeof(unsigned long long) * N);
  int* idx0 = (int*)alloc(sizeof(int) * 50000);
  int* idx1 = (int*)alloc(sizeof(int) * 25000);
  int* idx2 = (int*)alloc(sizeof(int) * 12500);
  int* newid = (int*)alloc(sizeof(int) * N);
  float* deg = (float*)alloc(sizeof(float) * 50000);
  const int MAXNB = (N + 255) / 256;
  unsigned int* hist = (unsigned int*)alloc(sizeof(unsigned int) * 256 * (size_t)MAXNB);

  auto cdiv = [](int a, int b) { return (a + b - 1) / b; };

  auto topk_sort = [&](const float* sc, int n, int k, int* idx_out) {
    int nb = cdiv(n, 256);
    build_keys<<<nb, 256, 0, stream>>>(sc, keyA, n);
    unsigned long long *src = keyA, *dst = keyB;
    for (int pass = 0; pass < 8; ++pass) {
      int shift = pass * 8;
      radix_count<<<nb, 256, 0, stream>>>(src, n, hist, nb, shift);
      radix_scan<<<1, 256, 0, stream>>>(hist, nb);
      radix_scatter<<<nb, 256, 0, stream>>>(src, dst, n, hist, nb, shift);
      unsigned long long* t = src; src = dst; dst = t;
    }
    extract_idx<<<cdiv(k, 256), 256, 0, stream>>>(src, idx_out, k);
  };

  float* hd[3]   = {HD0, HD1, HD2};
  int*   idxs[3] = {idx0, idx1, idx2};
  int*   rN[3]   = {row1, row2, row3};
  int*   cN[3]   = {col1, col2, col3};
  float* wN[3]   = {ewl1, ewl2, ewl3};

  // -------------------- down path --------------------
  const float* h_in = x;
  const int*   rp = erow0;
  const int*   cp = ecol0;
  const float* wp = ew0;
  for (int i = 0; i < L; ++i) {
    int n = sizes[i], k = sizes[i + 1];
    gemm64_wmma<<<cdiv(n, 16), 128, 0, stream>>>(h_in, down_w + (size_t)i * D * D,
                                                 down_b + (size_t)i * D, HW, n);
    fill_f32<<<cdiv(n * D, 256), 256, 0, stream>>>(ACC, 0.0f, n * D);
    spmm_atomic<<<cdiv(E, 8), 256, 0, stream>>>(rp, cp, wp, HW, ACC, E);
    elu_k<<<cdiv(n * D, 256), 256, 0, stream>>>(ACC, hd[i], n * D);
    // pool
    score_k<<<cdiv(n, 8), 256, 0, stream>>>(hd[i], proj_w + (size_t)i * D, proj_b, i,
                                            scores, n);
    topk_sort(scores, n, k, idxs[i]);
    fill_i32<<<cdiv(n, 256), 256, 0, stream>>>(newid, -1, n);
    set_newid<<<cdiv(k, 256), 256, 0, stream>>>(idxs[i], k, newid);
    fill_f32<<<cdiv(k, 256), 256, 0, stream>>>(deg, 0.0f, k);
    edge_remap<<<cdiv(E, 256), 256, 0, stream>>>(rp, cp, newid, E, rN[i], cN[i], wN[i], deg);
    ew_final<<<cdiv(E, 256), 256, 0, stream>>>(wN[i], rN[i], deg, E);
    gather_rows<<<cdiv(k * D, 256), 256, 0, stream>>>(hd[i], idxs[i], HCUR, k * D);
    h_in = HCUR; rp = rN[i]; cp = cN[i]; wp = wN[i];
  }

  // -------------------- bottom GCN (n = 12500) --------------------
  {
    int n = sizes[L];
    gemm64_wmma<<<cdiv(n, 16), 128, 0, stream>>>(HCUR, gcn_w, gcn_b, HW, n);
    fill_f32<<<cdiv(n * D, 256), 256, 0, stream>>>(ACC, 0.0f, n * D);
    spmm_atomic<<<cdiv(E, 8), 256, 0, stream>>>(row3, col3, ewl3, HW, ACC, E);
    elu_k<<<cdiv(n * D, 256), 256, 0, stream>>>(ACC, HCUR, n * D);
  }

  // -------------------- up path (results written straight into d_out) -----
  float* out0 = (float*)d_out;            // 25000 x 64
  float* out1 = out0 + (size_t)25000 * D; // 50000 x 64
  float* out2 = out1 + (size_t)50000 * D; // 100000 x 64
  float* out3 = out2 + (size_t)100000 * D;// 100000 x 64
  const int*   aR[3] = {erow0, row1, row2};
  const int*   aC[3] = {ecol0, col1, col2};
  const float* aW[3] = {ew0, ewl1, ewl2};
  float* outs_by_i[3] = {out2, out1, out0};

  const float* prev = HCUR;
  for (int j = 0; j < L; ++j) {
    int i = L - 1 - j;
    int n = sizes[i], k = sizes[i + 1];
    // unpool: ACC = zeros(n,D); ACC[idx] = prev
    fill_f32<<<cdiv(n * D, 256), 256, 0, stream>>>(ACC, 0.0f, n * D);
    scatter_rows<<<cdiv(k * D, 256), 256, 0, stream>>>(prev, idxs[i], ACC, k * D);
    gemm64_wmma<<<cdiv(n, 16), 128, 0, stream>>>(ACC, up_w + (size_t)i * D * D,
                                                 up_b + (size_t)i * D, HW, n);
    fill_f32<<<cdiv(n * D, 256), 256, 0, stream>>>(ACC, 0.0f, n * D);
    spmm_atomic<<<cdiv(E, 8), 256, 0, stream>>>(aR[i], aC[i], aW[i], HW, ACC, E);
    elu_add_k<<<cdiv(n * D, 256), 256, 0, stream>>>(ACC, hd[i], outs_by_i[i], n * D);
    prev = outs_by_i[i];
  }
  add_k<<<cdiv(N * D, 256), 256, 0, stream>>>(out2, x, out3, (int)((size_t)N * D));
}